// Net_16673063043527
// MI455X (gfx1250) — compile-verified
//
#include <hip/hip_runtime.h>

#define N_NODES 50000
#define F_IN    128
#define HID     256
#define CLS     40
#define NE      640000

typedef __attribute__((ext_vector_type(2))) float v2f;
typedef __attribute__((ext_vector_type(8))) float v8f;

// ---------------------------------------------------------------------------
// Zero a float region of the workspace (agg buffers / counts must start at 0).
// ---------------------------------------------------------------------------
__global__ void zero_f32(float* __restrict__ p, long long n) {
  long long i = (long long)blockIdx.x * blockDim.x + threadIdx.x;
  if (i < n) p[i] = 0.0f;
}

// ---------------------------------------------------------------------------
// Edge scatter: agg[dst] += feat[src] (float4-vectorized), plus in-degree
// counting on the first feature group. Uses native f32 global atomics.
// nfeat4 = F/4 (32 for layer 1, 64 for layer 2).
// ---------------------------------------------------------------------------
__global__ void scatter_accum(const float* __restrict__ feat,
                              const int*   __restrict__ src,
                              const int*   __restrict__ dst,
                              float* __restrict__ agg,
                              float* __restrict__ cnt,
                              int nfeat4, int do_cnt) {
  long long tid = (long long)blockIdx.x * blockDim.x + threadIdx.x;
  long long e = tid / nfeat4;
  int j = (int)(tid - e * nfeat4);
  if (e >= NE) return;
  int s = src[e];
  int d = dst[e];
  int F = nfeat4 * 4;
  float4 v = ((const float4*)feat)[(long long)s * nfeat4 + j];
  float* a = agg + (long long)d * F + j * 4;
  unsafeAtomicAdd(a + 0, v.x);
  unsafeAtomicAdd(a + 1, v.y);
  unsafeAtomicAdd(a + 2, v.z);
  unsafeAtomicAdd(a + 3, v.w);
  if (do_cnt && j == 0) unsafeAtomicAdd(cnt + d, 1.0f);
}

// ---------------------------------------------------------------------------
// agg[i] /= max(cnt(node),1): mean over in-neighbors.
// ---------------------------------------------------------------------------
__global__ void normalize_mean(float* __restrict__ agg,
                               const float* __restrict__ cnt, int F) {
  long long i = (long long)blockIdx.x * blockDim.x + threadIdx.x;
  long long n = (long long)N_NODES * F;
  if (i >= n) return;
  int node = (int)(i / F);
  float c = cnt[node];
  c = c > 1.0f ? c : 1.0f;
  agg[i] = agg[i] / c;
}

// ---------------------------------------------------------------------------
// Dual GEMM via V_WMMA_F32_16X16X4_F32:
//   out[M, ncols] = A1 @ W1^T + A2 @ W2^T + bias   (optional ReLU)
// A1/A2: [M, K] row-major.  W1/W2: [ncols, K] row-major.  M = N_NODES (=3125*16).
// One wave owns one 16x16 output tile. A fragment (16x4, 2 VGPR):
//   lanes 0-15 -> M=lane, K=k0/k0+1 ; lanes 16-31 -> M=lane-16, K=k0+2/k0+3.
// B fragment (4x16) symmetric with N on lanes. C/D: lane -> N, vgpr -> M(+8).
//
// Out-of-range output columns (ncols=40 case): the B row pointer is CLAMPED
// (branch-free) instead of masked — a lane's B column only feeds the D column
// held by the same lane, which is guarded at the store, so garbage is harmless.
// This keeps the inner loop at 2 x global_load_b64 + 1 x v_wmma, no EXEC ops.
// ---------------------------------------------------------------------------
__global__ void dual_gemm_wmma(const float* __restrict__ A1,
                               const float* __restrict__ W1,
                               const float* __restrict__ A2,
                               const float* __restrict__ W2,
                               const float* __restrict__ bias,
                               float* __restrict__ out,
                               int K, int ncols, int ldo, int do_relu) {
  const int lane   = threadIdx.x & 31;
  const int wid    = blockIdx.x * (blockDim.x >> 5) + (threadIdx.x >> 5);
  const int ctiles = (ncols + 15) >> 4;
  const int mtiles = N_NODES >> 4;
  if (wid >= mtiles * ctiles) return;          // whole-wave exit: EXEC stays full below
  const int rt = wid / ctiles;
  const int ct = wid - rt * ctiles;

  const int mrow  = rt * 16 + (lane & 15);     // A row held by this lane
  const int ncol  = ct * 16 + (lane & 15);     // B col held by this lane
  const int khalf = (lane >> 4) * 2;           // lane's K sub-offset within group-of-4
  const bool cvalid  = (ncol < ncols);
  const int  ncol_ld = cvalid ? ncol : (ncols - 1);   // clamped load column

  // 32-bit offsets (max 50000*256 < 2^31) off pre-biased pointers.
  const float* arow1 = A1 + (unsigned)mrow    * (unsigned)K + khalf;
  const float* brow1 = W1 + (unsigned)ncol_ld * (unsigned)K + khalf;
  const float* arow2 = A2 + (unsigned)mrow    * (unsigned)K + khalf;
  const float* brow2 = W2 + (unsigned)ncol_ld * (unsigned)K + khalf;

  v8f c = {};

  // Pass 1: A1 x W1^T
  #pragma unroll 8
  for (int k0 = 0; k0 < K; k0 += 4) {
    float2 av = *(const float2*)(arow1 + k0);
    float2 bv = *(const float2*)(brow1 + k0);
    v2f a; a.x = av.x; a.y = av.y;
    v2f b; b.x = bv.x; b.y = bv.y;
    c = __builtin_amdgcn_wmma_f32_16x16x4_f32(false, a, false, b,
                                              (short)0, c, false, false);
  }
  // Pass 2: A2 x W2^T (accumulates into same C)
  #pragma unroll 8
  for (int k0 = 0; k0 < K; k0 += 4) {
    float2 av = *(const float2*)(arow2 + k0);
    float2 bv = *(const float2*)(brow2 + k0);
    v2f a; a.x = av.x; a.y = av.y;
    v2f b; b.x = bv.x; b.y = bv.y;
    c = __builtin_amdgcn_wmma_f32_16x16x4_f32(false, a, false, b,
                                              (short)0, c, false, false);
  }

  // Epilogue: bias (+ReLU) and store. C/D layout: lane -> N; vgpr i -> M = i (+8).
  if (cvalid) {
    float bv = bias[ncol];
    unsigned rbase = (unsigned)(rt * 16 + ((lane < 16) ? 0 : 8));
    float* orow = out + (unsigned)rbase * (unsigned)ldo + ncol;
    #pragma unroll
    for (int i = 0; i < 8; ++i) {
      float v = c[i] + bv;
      if (do_relu) v = v > 0.0f ? v : 0.0f;
      orow[(unsigned)i * (unsigned)ldo] = v;
    }
  }
}

// ---------------------------------------------------------------------------
// Row-wise log_softmax over CLS=40 logits. One thread per row.
// ---------------------------------------------------------------------------
__global__ void log_softmax_rows(const float* __restrict__ logits,
                                 float* __restrict__ out) {
  int r = blockIdx.x * blockDim.x + threadIdx.x;
  if (r >= N_NODES) return;
  const float* l = logits + (long long)r * CLS;
  float m = -3.402823466e38f;
  #pragma unroll
  for (int j = 0; j < CLS; ++j) m = fmaxf(m, l[j]);
  float s = 0.0f;
  #pragma unroll
  for (int j = 0; j < CLS; ++j) s += expf(l[j] - m);
  float ls = logf(s);
  float* o = out + (long long)r * CLS;
  #pragma unroll
  for (int j = 0; j < CLS; ++j) o[j] = l[j] - m - ls;
}

// ---------------------------------------------------------------------------
extern "C" void kernel_launch(void* const* d_in, const int* in_sizes, int n_in,
                              void* d_out, int out_size, void* d_ws, size_t ws_size,
                              hipStream_t stream) {
  (void)in_sizes; (void)n_in; (void)out_size; (void)ws_size;

  const float* x    = (const float*)d_in[0];
  const int*   ei   = (const int*)  d_in[1];   // [2, E]: src then dst
  const float* W1_l = (const float*)d_in[2];
  const float* b1   = (const float*)d_in[3];
  const float* W1_r = (const float*)d_in[4];
  const float* W2_l = (const float*)d_in[5];
  const float* b2   = (const float*)d_in[6];
  const float* W2_r = (const float*)d_in[7];
  const int* src = ei;
  const int* dst = ei + NE;
  float* out = (float*)d_out;

  // Workspace layout (floats): cnt | agg1 | h | agg2 | logits
  float* ws     = (float*)d_ws;
  float* cnt    = ws;                                    // N
  float* agg1   = cnt  + N_NODES;                        // N * F_IN
  float* h      = agg1 + (long long)N_NODES * F_IN;      // N * HID
  float* agg2   = h    + (long long)N_NODES * HID;       // N * HID
  float* logits = agg2 + (long long)N_NODES * HID;       // N * CLS

  const int TB = 256;
  auto blocks = [](long long n, int tb) { return (unsigned)((n + tb - 1) / tb); };

  // ---- Layer 1 ----
  long long nz1 = (long long)N_NODES * (1 + F_IN);       // cnt + agg1 contiguous
  zero_f32<<<blocks(nz1, TB), TB, 0, stream>>>(cnt, nz1);

  long long sc1 = (long long)NE * (F_IN / 4);
  scatter_accum<<<blocks(sc1, TB), TB, 0, stream>>>(x, src, dst, agg1, cnt,
                                                    F_IN / 4, /*do_cnt=*/1);

  long long nm1 = (long long)N_NODES * F_IN;
  normalize_mean<<<blocks(nm1, TB), TB, 0, stream>>>(agg1, cnt, F_IN);

  // h = relu(agg1 @ W1_l^T + x @ W1_r^T + b1), [N, 256]
  {
    int tiles = (N_NODES / 16) * (HID / 16);             // 3125 * 16
    dual_gemm_wmma<<<blocks(tiles, 8), TB, 0, stream>>>(
        agg1, W1_l, x, W1_r, b1, h, F_IN, HID, HID, /*relu=*/1);
  }

  // ---- Layer 2 ----
  long long nz2 = (long long)N_NODES * HID;
  zero_f32<<<blocks(nz2, TB), TB, 0, stream>>>(agg2, nz2);

  long long sc2 = (long long)NE * (HID / 4);
  scatter_accum<<<blocks(sc2, TB), TB, 0, stream>>>(h, src, dst, agg2, cnt,
                                                    HID / 4, /*do_cnt=*/0);

  long long nm2 = (long long)N_NODES * HID;
  normalize_mean<<<blocks(nm2, TB), TB, 0, stream>>>(agg2, cnt, HID);

  // logits = agg2 @ W2_l^T + h @ W2_r^T + b2, [N, 40]
  {
    int ctiles = (CLS + 15) / 16;                        // 3
    int tiles = (N_NODES / 16) * ctiles;                 // 9375
    dual_gemm_wmma<<<blocks(tiles, 8), TB, 0, stream>>>(
        agg2, W2_l, h, W2_r, b2, logits, HID, CLS, CLS, /*relu=*/0);
  }

  // ---- log_softmax ----
  log_softmax_rows<<<blocks(N_NODES, TB), TB, 0, stream>>>(logits, out);
}